// GCN2_6133213299120
// MI455X (gfx1250) — compile-verified
//
#include <hip/hip_runtime.h>
#include <math.h>

// ---------------- problem constants (match reference) ----------------
#define N_NODES 100000
#define IN_DIM  512
#define HID     256

typedef __attribute__((ext_vector_type(2))) float v2f;
typedef __attribute__((ext_vector_type(8))) float v8f;
typedef __attribute__((ext_vector_type(4))) int   v4i;

// CDNA5 async global->LDS path (ASYNCcnt), with sync fallback if the
// toolchain doesn't declare the builtins.
#if __has_builtin(__builtin_amdgcn_global_load_async_to_lds_b128) && \
    __has_builtin(__builtin_amdgcn_s_wait_asynccnt)
#define HAVE_ASYNC_LDS 1
typedef __attribute__((address_space(1))) v4i* gv4p;   // global int4*
typedef __attribute__((address_space(3))) v4i* lv4p;   // LDS int4*
#else
#define HAVE_ASYNC_LDS 0
#endif

// ---------------- WMMA f32 GEMM:  Y[N,HID] = X[N,K] @ W[K,HID] -------
// Block = 256 threads (8 waves). Block tile: BM=128 rows x BN=64 cols.
// Double-buffered LDS; A tile staged via async global->LDS B128 copies,
// W tile software-pipelined through registers and stored k-paired so the
// B fragment is a single aligned ds_load_b64 (no v_mov repacking).
#define BM 128
#define BN 64
#define BK 32
#define AS_STRIDE 36   // floats; mult of 4 (16B chunks), conflict-free A-frag reads
#define WS2_STRIDE 80  // float2 units; 160 floats == 32 mod 64 banks between kp rows

__global__ __launch_bounds__(256)
void gemm_wmma_f32(const float* __restrict__ X, const float* __restrict__ W,
                   float* __restrict__ Y, int N, int K) {
    __shared__ float  As[2][BM * AS_STRIDE];            // 2 x 18,432 B
    __shared__ float2 Ws2[2][(BK / 2) * WS2_STRIDE];    // 2 x 10,240 B

    const int tid     = threadIdx.x;
    const int wave    = tid >> 5;
    const int lane    = tid & 31;
    const int lhalf   = lane >> 4;   // 0: lanes 0-15, 1: lanes 16-31
    const int l16     = lane & 15;
    const int rowBase = blockIdx.x * BM;
    const int colBase = blockIdx.y * BN;
    const int nchunks = K / BK;

    v8f acc[4] = {};

    // ---- A tile stager: 4x B128 per thread, per-lane clamp for tail rows ----
    auto stage_A = [&](float* dst, int k0) {
        #pragma unroll
        for (int i = 0; i < 4; ++i) {
            int idx  = tid + i * 256;         // 1024 x 16B
            int r    = idx >> 3;              // 8 chunks per row
            int kq   = idx & 7;
            int grow = rowBase + r;
            if (grow >= N) grow = N - 1;      // clamp: rows never stored
            const float* g = X + (size_t)grow * K + k0 + kq * 4;
            float*       l = dst + r * AS_STRIDE + kq * 4;
#if HAVE_ASYNC_LDS
            __builtin_amdgcn_global_load_async_to_lds_b128((gv4p)g, (lv4p)l, 0, 0);
#else
            *(float4*)l = *(const float4*)g;
#endif
        }
    };
    auto wait_A = [&]() {
#if HAVE_ASYNC_LDS
        __builtin_amdgcn_s_wait_asynccnt(0);
#endif
    };
    // ---- W tile: 2 float4 per thread, register-pipelined ----
    auto load_W = [&](int k0, float4& w0, float4& w1) {
        #pragma unroll
        for (int i = 0; i < 2; ++i) {
            int idx = tid + i * 256;          // 512 x 16B
            int r   = idx >> 4;               // 16 chunks per row
            int nq  = idx & 15;
            float4 v = *(const float4*)(W + (size_t)(k0 + r) * HID + colBase + nq * 4);
            if (i == 0) w0 = v; else w1 = v;
        }
    };
    auto store_W = [&](float2* dst, float4 w0, float4 w1) {
        #pragma unroll
        for (int i = 0; i < 2; ++i) {
            int idx = tid + i * 256;
            int r   = idx >> 4;
            int nq  = idx & 15;
            float4 v = (i == 0) ? w0 : w1;
            // pack rows (2k, 2k+1) as float2 so B frag is one ds_load_b64
            float* wp = (float*)dst + (r >> 1) * (WS2_STRIDE * 2) + nq * 8 + (r & 1);
            wp[0] = v.x; wp[2] = v.y; wp[4] = v.z; wp[6] = v.w;
        }
    };

    // ---- prologue: stage chunk 0 ----
    float4 w0, w1;
    stage_A(As[0], 0);
    load_W(0, w0, w1);
    store_W(Ws2[0], w0, w1);
    wait_A();
    __syncthreads();

    for (int c = 0; c < nchunks; ++c) {
        const int buf = c & 1;
        if (c + 1 < nchunks) {
            stage_A(As[buf ^ 1], (c + 1) * BK);     // async into other buffer
            load_W((c + 1) * BK, w0, w1);           // regs for other buffer
        }

        const float*  Ab = As[buf];
        const float2* Wb = Ws2[buf];
        #pragma unroll
        for (int ks = 0; ks < BK; ks += 4) {
            // A frag (16x4): lanes 0-15 hold K=ks,ks+1 ; lanes 16-31 K=ks+2,ks+3
            const int ka = ks + 2 * lhalf;
            v2f a;
            a.x = Ab[(wave * 16 + l16) * AS_STRIDE + ka];
            a.y = Ab[(wave * 16 + l16) * AS_STRIDE + ka + 1];
            const int kp = (ks >> 1) + lhalf;       // paired-row index
            #pragma unroll
            for (int t = 0; t < 4; ++t) {
                v2f b = *(const v2f*)&Wb[kp * WS2_STRIDE + t * 16 + l16];
                acc[t] = __builtin_amdgcn_wmma_f32_16x16x4_f32(
                    false, a, false, b, (short)0, acc[t], false, false);
            }
        }
        __syncthreads();                            // all reads of buf done

        if (c + 1 < nchunks) {
            store_W(Ws2[buf ^ 1], w0, w1);
            wait_A();                               // async A landed
            __syncthreads();                        // visible to all waves
        }
    }

    // ---- store: C/D 16x16 layout: VGPR r -> (M=r, N=lane) / (M=r+8, N=lane-16)
    #pragma unroll
    for (int t = 0; t < 4; ++t) {
        #pragma unroll
        for (int r = 0; r < 8; ++r) {
            int row = rowBase + wave * 16 + r + 8 * lhalf;
            int col = colBase + t * 16 + l16;
            if (row < N)
                Y[(size_t)row * HID + col] = acc[t][r];
        }
    }
}

// ---------------- SpMM scatter: dst[rows[e]] += vals[e] * src[cols[e]] ----
// One wave per edge; lane handles 8 of the 256 features (2x float4 loads,
// 8 agent-scope relaxed f32 atomic adds). dst is L2-resident (102 MB).
__global__ __launch_bounds__(256)
void spmm_scatter(const int* __restrict__ rows, const int* __restrict__ cols,
                  const float* __restrict__ vals, const float* __restrict__ src,
                  float* __restrict__ dst, int n_edges) {
    int wid = blockIdx.x * (blockDim.x >> 5) + (threadIdx.x >> 5);
    if (wid >= n_edges) return;
    int lane = threadIdx.x & 31;
    int r = rows[wid];
    int c = cols[wid];
    float v = vals[wid];
    const float4* s = (const float4*)(src + (size_t)c * HID);
    float* d = dst + (size_t)r * HID;
    #pragma unroll
    for (int i = 0; i < 2; ++i) {
        float4 x = s[lane + i * 32];
        int f = (lane + i * 32) * 4;
        __hip_atomic_fetch_add(d + f + 0, v * x.x, __ATOMIC_RELAXED, __HIP_MEMORY_SCOPE_AGENT);
        __hip_atomic_fetch_add(d + f + 1, v * x.y, __ATOMIC_RELAXED, __HIP_MEMORY_SCOPE_AGENT);
        __hip_atomic_fetch_add(d + f + 2, v * x.z, __ATOMIC_RELAXED, __HIP_MEMORY_SCOPE_AGENT);
        __hip_atomic_fetch_add(d + f + 3, v * x.w, __ATOMIC_RELAXED, __HIP_MEMORY_SCOPE_AGENT);
    }
}

// ---------------- bias + relu:  out = max(acc + b[col], 0) ----------------
__global__ __launch_bounds__(256)
void bias_relu(const float* __restrict__ acc, const float* __restrict__ b,
               float* __restrict__ out, int n_elems) {
    int idx = blockIdx.x * blockDim.x + threadIdx.x;
    if (idx >= n_elems) return;
    float x = acc[idx] + b[idx & (HID - 1)];
    out[idx] = fmaxf(x, 0.f);
}

// ---------------- head: (logits+b2) @ fc1 -> relu -> fc2 -> log_softmax ---
// One wave per node; lane j owns h_a[j] (32-wide), shuffle-reduce for fc2.
__global__ __launch_bounds__(256)
void head_kernel(const float* __restrict__ acc2, const float* __restrict__ b2,
                 const float* __restrict__ fc1W,  // [256][32]
                 const float* __restrict__ fc1b,  // [32]
                 const float* __restrict__ fc2W,  // [32][2]
                 const float* __restrict__ fc2b,  // [2]
                 float* __restrict__ out, int n) {
    int node = blockIdx.x * (blockDim.x >> 5) + (threadIdx.x >> 5);
    if (node >= n) return;
    int lane = threadIdx.x & 31;
    const float* lrow = acc2 + (size_t)node * HID;

    float s = 0.f;
    #pragma unroll 8
    for (int k = 0; k < HID; ++k) {
        float lk = lrow[k] + b2[k];          // wave-uniform loads
        s = fmaf(lk, fc1W[k * 32 + lane], s);
    }
    s += fc1b[lane];
    s = fmaxf(s, 0.f);                       // h_a[lane]

    float p0 = s * fc2W[lane * 2 + 0];
    float p1 = s * fc2W[lane * 2 + 1];
    #pragma unroll
    for (int off = 16; off > 0; off >>= 1) {
        p0 += __shfl_down(p0, off, 32);
        p1 += __shfl_down(p1, off, 32);
    }
    if (lane == 0) {
        p0 += fc2b[0];
        p1 += fc2b[1];
        float m  = fmaxf(p0, p1);
        float mn = fminf(p0, p1);
        float lse = m + log1pf(expf(mn - m));
        out[(size_t)node * 2 + 0] = p0 - lse;
        out[(size_t)node * 2 + 1] = p1 - lse;
    }
}

// --------------------------------------------------------------------------
extern "C" void kernel_launch(void* const* d_in, const int* in_sizes, int n_in,
                              void* d_out, int out_size, void* d_ws, size_t ws_size,
                              hipStream_t stream) {
    const float* X     = (const float*)d_in[0];
    const int*   erows = (const int*)  d_in[1];
    const int*   ecols = (const int*)  d_in[2];
    const float* evals = (const float*)d_in[3];
    const float* W1    = (const float*)d_in[4];
    const float* b1    = (const float*)d_in[5];
    const float* W2    = (const float*)d_in[6];
    const float* b2    = (const float*)d_in[7];
    const float* fc1W  = (const float*)d_in[8];
    const float* fc1b  = (const float*)d_in[9];
    const float* fc2W  = (const float*)d_in[10];
    const float* fc2b  = (const float*)d_in[11];
    float* out = (float*)d_out;

    const int n_edges = in_sizes[1];
    const size_t BUF = (size_t)N_NODES * HID;           // 25.6M floats
    float* bufA = (float*)d_ws;                         // 102.4 MB
    float* bufB = bufA + BUF;                           // 102.4 MB

    dim3 gemmGrid((N_NODES + BM - 1) / BM, HID / BN);   // (782, 4)
    int spmmBlocks = (n_edges + 7) / 8;                 // 8 waves/block, 1 edge/wave
    int eltBlocks  = (int)((BUF + 255) / 256);
    int headBlocks = (N_NODES + 7) / 8;

    // ---- layer 1: support1 = X @ W1 ----
    gemm_wmma_f32<<<gemmGrid, 256, 0, stream>>>(X, W1, bufA, N_NODES, IN_DIM);
    // ---- spmm1: bufB = A @ support1 ----
    (void)hipMemsetAsync(bufB, 0, BUF * sizeof(float), stream);
    spmm_scatter<<<spmmBlocks, 256, 0, stream>>>(erows, ecols, evals, bufA, bufB, n_edges);
    // ---- h = relu(bufB + b1) -> bufA ----
    bias_relu<<<eltBlocks, 256, 0, stream>>>(bufB, b1, bufA, (int)BUF);
    // ---- layer 2: support2 = h @ W2 -> bufB ----
    gemm_wmma_f32<<<gemmGrid, 256, 0, stream>>>(bufA, W2, bufB, N_NODES, HID);
    // ---- spmm2: bufA = A @ support2 ----
    (void)hipMemsetAsync(bufA, 0, BUF * sizeof(float), stream);
    spmm_scatter<<<spmmBlocks, 256, 0, stream>>>(erows, ecols, evals, bufB, bufA, n_edges);
    // ---- head: log_softmax(relu((bufA + b2) @ fc1) @ fc2) -> out ----
    head_kernel<<<headBlocks, 256, 0, stream>>>(bufA, b2, fc1W, fc1b, fc2W, fc2b,
                                                out, N_NODES);
}